// JPEG_3281355014532
// MI455X (gfx1250) — compile-verified
//
#include <hip/hip_runtime.h>

typedef __attribute__((ext_vector_type(2))) float v2f;
typedef __attribute__((ext_vector_type(8))) float v8f;

// Reciprocal luminance quantization table: 100 / LUM[i]  (Q = LUM/100, we multiply by 1/Q)
__constant__ float RQ[64] = {
    100.0f/16.0f, 100.0f/11.0f, 100.0f/10.0f, 100.0f/16.0f, 100.0f/24.0f, 100.0f/40.0f,  100.0f/51.0f,  100.0f/61.0f,
    100.0f/12.0f, 100.0f/12.0f, 100.0f/14.0f, 100.0f/19.0f, 100.0f/26.0f, 100.0f/58.0f,  100.0f/60.0f,  100.0f/55.0f,
    100.0f/14.0f, 100.0f/13.0f, 100.0f/16.0f, 100.0f/24.0f, 100.0f/40.0f, 100.0f/57.0f,  100.0f/69.0f,  100.0f/56.0f,
    100.0f/14.0f, 100.0f/17.0f, 100.0f/22.0f, 100.0f/29.0f, 100.0f/51.0f, 100.0f/87.0f,  100.0f/80.0f,  100.0f/62.0f,
    100.0f/18.0f, 100.0f/22.0f, 100.0f/37.0f, 100.0f/56.0f, 100.0f/68.0f, 100.0f/109.0f, 100.0f/103.0f, 100.0f/77.0f,
    100.0f/24.0f, 100.0f/36.0f, 100.0f/55.0f, 100.0f/64.0f, 100.0f/81.0f, 100.0f/104.0f, 100.0f/113.0f, 100.0f/92.0f,
    100.0f/49.0f, 100.0f/64.0f, 100.0f/78.0f, 100.0f/87.0f, 100.0f/103.0f,100.0f/121.0f, 100.0f/120.0f, 100.0f/101.0f,
    100.0f/72.0f, 100.0f/92.0f, 100.0f/95.0f, 100.0f/98.0f, 100.0f/112.0f,100.0f/100.0f, 100.0f/103.0f, 100.0f/99.0f
};

static __device__ inline v8f wmma_f32_k4(v2f a, v2f b, v8f c) {
    // D = A(16x4 f32) x B(4x16 f32) + C(16x16 f32)
    return __builtin_amdgcn_wmma_f32_16x16x4_f32(
        /*neg_a=*/false, a, /*neg_b=*/false, b,
        /*c_mod=*/(short)0, c, /*reuse_a=*/false, /*reuse_b=*/false);
}

// One wave processes one 16x16 pixel tile = four 8x8 JPEG blocks.
// Stage 1: T = Chat * X            (Chat = blockdiag(C8, C8), X loaded in B-layout)
// Stage 2: W = Chat * T^T = Z^T    (T^T via padded-stride LDS, Z = Chat*X*Chat^T)
// Scale:   out = Z / (factor * Q)  (Z^T handled purely in store addressing)
__global__ __launch_bounds__(256) void jpeg_dct_q_kernel(
    const float* __restrict__ img, const float* __restrict__ qf,
    float* __restrict__ out, int n_tiles)
{
    __shared__ float lds[8][16 * 17];   // per-wave 16x16 tile, stride 17 dwords (bank-conflict-free transpose)

    const int lane = threadIdx.x & 31;
    const int wid  = threadIdx.x >> 5;
    const int tile = blockIdx.x * 8 + wid;
    if (tile >= n_tiles) return;        // wave-uniform; EXEC stays all-ones for WMMA

    const int imgIdx = tile >> 12;      // 64*64 tiles per 1024x1024 image
    const int t      = tile & 4095;
    const int th     = t >> 6;          // tile row  (0..63)
    const int tw     = t & 63;          // tile col  (0..63)

    // ---- A operand: Chat = blockdiag(C8, C8), 4 K-chunks x 2 VGPRs ----
    // A layout (32-bit 16x4): lane<16 -> (M=lane, K=k0+j), lane>=16 -> (M=lane-16, K=k0+2+j)
    const int   m    = lane & 15;       // output frequency row
    const int   mi   = m & 7;
    const float fm   = (mi == 0) ? 0.35355339059327373f : 0.5f;
    const int   rsel = (lane >> 4) << 1;  // 0 for low half, 2 for high half
    float amat[8];
#pragma unroll
    for (int c = 0; c < 4; ++c) {
#pragma unroll
        for (int j = 0; j < 2; ++j) {
            const int k = 4 * c + rsel + j;    // spatial column of Chat
            float v = 0.0f;
            if ((m >> 3) == (k >> 3))          // block-diagonal
                v = fm * __cosf(0.19634954084936207f * (float)(2 * (k & 7) + 1) * (float)mi);
            amat[2 * c + j] = v;
        }
    }

    // ---- Load X (16x16 pixels) directly into B-operand layout, fold -128 ----
    // B layout (32-bit 4x16): lane<16 -> (K=k0+j, N=lane), lane>=16 -> (K=k0+2+j, N=lane-16)
    const int col = lane & 15;
    const float* base = img + ((long)imgIdx * 1024 + th * 16) * 1024 + tw * 16 + col;
    float xb[8];
#pragma unroll
    for (int c = 0; c < 4; ++c) {
#pragma unroll
        for (int j = 0; j < 2; ++j) {
            const int row = 4 * c + rsel + j;
            xb[2 * c + j] = base[(long)row * 1024] - 128.0f;   // coalesced 64B segments
        }
    }

    // ---- Stage 1: T = Chat * X  (4 chained K=4 WMMAs) ----
    v8f acc = {0.f, 0.f, 0.f, 0.f, 0.f, 0.f, 0.f, 0.f};
#pragma unroll
    for (int c = 0; c < 4; ++c) {
        v2f a = {amat[2 * c], amat[2 * c + 1]};
        v2f b = {xb[2 * c],   xb[2 * c + 1]};
        acc = wmma_f32_k4(a, b, acc);
    }

    // ---- Transpose T through LDS (stride-17 padding) ----
    // D layout: VGPR r: lane<16 -> T[r][lane], lane>=16 -> T[r+8][lane-16]
    float* sm = lds[wid];
    const int rowoff = (lane >> 4) << 3;   // 0 or 8
#pragma unroll
    for (int r = 0; r < 8; ++r)
        sm[(r + rowoff) * 17 + col] = acc[r];

    // Same-wave LDS ops are in-order; wait for stores before dependent loads.
    asm volatile("s_wait_dscnt 0" ::: "memory");

    // Read T^T in B layout: T^T[k][n] = T[n][k] at offset n*17 + k (conflict-free banks)
    float tb[8];
#pragma unroll
    for (int c = 0; c < 4; ++c) {
#pragma unroll
        for (int j = 0; j < 2; ++j) {
            const int k = 4 * c + rsel + j;
            tb[2 * c + j] = sm[col * 17 + k];
        }
    }

    // ---- Stage 2: W = Chat * T^T = Z^T ----
    v8f w = {0.f, 0.f, 0.f, 0.f, 0.f, 0.f, 0.f, 0.f};
#pragma unroll
    for (int c = 0; c < 4; ++c) {
        v2f a = {amat[2 * c], amat[2 * c + 1]};
        v2f b = {tb[2 * c],   tb[2 * c + 1]};
        w = wmma_f32_k4(a, b, w);
    }

    // ---- Quantizer scale + store ----
    // W VGPR r, lane l:  Z[n16][m16] with n16 = (l&15), m16 = r + 8*(l>>4)
    //   row-freq k = l&7, col-freq = r, block-row a = (l>>3)&1, block-col b = l>>4
    // 1/factor computed branch-free:
    //   q < 50  : factor = 5000/q      -> 1/factor = q / 5000 (exact, no division)
    //   q >= 50 : factor = 200 - 2q    -> 1/factor = rcp(200 - 2q)
    const float q    = qf[imgIdx];
    const float invA = q * 0.0002f;
    const float invB = __frcp_rn(200.0f - 2.0f * q);
    const float invf = (q < 50.0f) ? invA : invB;    // v_cndmask, no exec divergence

    const int kf = lane & 7;
    const int a_ = (lane >> 3) & 1;
    const int b_ = lane >> 4;
    const int hb = th * 2 + a_;
    const int wb = tw * 2 + b_;
    float* obase = out + (((long)imgIdx * 64 + kf * 8) * 128 + hb) * 128 + wb;
#pragma unroll
    for (int r = 0; r < 8; ++r)                      // channel = kf*8 + r
        obase[(long)r * 16384] = w[r] * invf * RQ[kf * 8 + r];
}

extern "C" void kernel_launch(void* const* d_in, const int* in_sizes, int n_in,
                              void* d_out, int out_size, void* d_ws, size_t ws_size,
                              hipStream_t stream) {
    const float* image = (const float*)d_in[0];   // (B,1,1024,1024) fp32
    const float* qf    = (const float*)d_in[1];   // (B,) fp32
    float* out         = (float*)d_out;           // (B,64,128,128) fp32

    const int images  = in_sizes[1];
    const int n_tiles = images * 64 * 64;         // 16x16 tiles
    const int blocks  = (n_tiles + 7) / 8;        // 8 waves (tiles) per 256-thread block

    hipLaunchKernelGGL(jpeg_dct_q_kernel, dim3(blocks), dim3(256), 0, stream,
                       image, qf, out, n_tiles);
}